// MIL_Pooling_16844861735558
// MI455X (gfx1250) — compile-verified
//
#include <hip/hip_runtime.h>
#include <math.h>

// ---------------------------------------------------------------------------
// Kernel 1: exclusive prefix sum of bag sizes -> starts[0..n], starts[n]=TOTAL
// One block of 1024 threads, each handling a contiguous chunk of 16 sizes.
// Fast path: 4 x int4 (b128) unguarded loads when the whole chunk is in range
// (true for n_bags = 16384), avoiding 16 exec-mask branches per thread.
// ---------------------------------------------------------------------------
__global__ __launch_bounds__(1024) void scan_kernel(const int* __restrict__ sizes,
                                                    int* __restrict__ starts,
                                                    int n) {
    __shared__ int sums[1024];
    const int t = threadIdx.x;
    const int PER = 16;                 // 1024 * 16 = 16384 bags
    const int base = t * PER;

    int vals[PER];
    if (base + PER <= n) {
        const int4* p = (const int4*)(sizes + base);
        int4 a = p[0], b = p[1], c = p[2], d = p[3];
        vals[0]=a.x;  vals[1]=a.y;  vals[2]=a.z;  vals[3]=a.w;
        vals[4]=b.x;  vals[5]=b.y;  vals[6]=b.z;  vals[7]=b.w;
        vals[8]=c.x;  vals[9]=c.y;  vals[10]=c.z; vals[11]=c.w;
        vals[12]=d.x; vals[13]=d.y; vals[14]=d.z; vals[15]=d.w;
    } else {
        #pragma unroll
        for (int i = 0; i < PER; ++i)
            vals[i] = (base + i < n) ? sizes[base + i] : 0;
    }

    int local[PER];
    int s = 0;
    #pragma unroll
    for (int i = 0; i < PER; ++i) { local[i] = s; s += vals[i]; }

    sums[t] = s;
    __syncthreads();

    // Hillis-Steele inclusive scan over the 1024 chunk totals
    for (int off = 1; off < 1024; off <<= 1) {
        int add = (t >= off) ? sums[t - off] : 0;
        __syncthreads();
        sums[t] += add;
        __syncthreads();
    }

    const int prefix = (t == 0) ? 0 : sums[t - 1];
    #pragma unroll
    for (int i = 0; i < PER; ++i)
        if (base + i < n) starts[base + i] = prefix + local[i];
    if (t == 1023) starts[n] = sums[1023];   // == TOTAL
}

// ---------------------------------------------------------------------------
// Kernel 2: ragged segment-max, one bag per block, 256 threads (8 wave32).
// Rows staged 16 at a time into double-buffered LDS via async-to-LDS b128
// (ASYNCcnt path, gfx1250), non-temporal (stream is 256 MB > 192 MB L2 and
// read exactly once). Each thread copies one 16-byte segment per chunk:
//   seg  = t & 15   (which 16B of the 256B row)
//   rowi = t >> 4   (which of the 16 rows in the chunk)
// Lanes past the bag end clamp to the last valid row so every wave issues
// exactly one async op per chunk -> per-wave ASYNCcnt bookkeeping is exact.
// Consume mapping: feat = t & 63, rowgroup = t >> 6 (bank-conflict-free:
// lanes within a wave read consecutive LDS words across the 64 banks).
// ---------------------------------------------------------------------------
#define CHUNK_ROWS 16
#define KDIM 64

__global__ __launch_bounds__(256) void pool_kernel(const float* __restrict__ x,
                                                   const int* __restrict__ starts,
                                                   float* __restrict__ out) {
    __shared__ float buf[2][CHUNK_ROWS * KDIM];   // 2 x 4 KB staging tiles
    __shared__ float red[256];                    // cross-rowgroup reduce

    const int bag = blockIdx.x;
    const int t   = threadIdx.x;
    const int start = starts[bag];
    const int end   = starts[bag + 1];
    const int n     = end - start;                // >= 1 by construction
    const int nch   = (n + CHUNK_ROWS - 1) / CHUNK_ROWS;

    // LDS byte offsets of the two staging buffers (generic ptr low 32 bits
    // == addrspace(3) offset on AMDGPU).
    const unsigned lds0 = (unsigned)(size_t)(&buf[0][0]);
    const unsigned lds1 = (unsigned)(size_t)(&buf[1][0]);

    const int seg  = t & 15;        // 16B segment within a row
    const int rowi = t >> 4;        // row within the chunk
    const unsigned long long xbase = (unsigned long long)(size_t)x;
    const unsigned lane_lds = (unsigned)(rowi * 256 + seg * 16);

    // Prologue: issue chunk 0 into buffer 0
    {
        int grow = start + rowi;
        if (grow > end - 1) grow = end - 1;       // clamp (keeps EXEC full)
        unsigned long long ga = xbase + (unsigned long long)grow * 256ull
                                      + (unsigned long long)(seg * 16);
        unsigned la = lds0 + lane_lds;
        asm volatile("global_load_async_to_lds_b128 %0, %1, off th:TH_LOAD_NT"
                     :: "v"(la), "v"(ga) : "memory");
    }

    const int feat = t & 63;
    const int rg   = t >> 6;        // 4 row-groups
    float m = -INFINITY;

    for (int c = 0; c < nch; ++c) {
        if (c + 1 < nch) {
            // issue next chunk into the other buffer, then wait for current
            int grow = start + (c + 1) * CHUNK_ROWS + rowi;
            if (grow > end - 1) grow = end - 1;
            unsigned long long ga = xbase + (unsigned long long)grow * 256ull
                                          + (unsigned long long)(seg * 16);
            unsigned la = (((c + 1) & 1) ? lds1 : lds0) + lane_lds;
            asm volatile("global_load_async_to_lds_b128 %0, %1, off th:TH_LOAD_NT"
                         :: "v"(la), "v"(ga) : "memory");
            asm volatile("s_wait_asynccnt 0x1" ::: "memory");
        } else {
            asm volatile("s_wait_asynccnt 0x0" ::: "memory");
        }
        __syncthreads();            // chunk c fully landed for all waves

        const float* b = buf[c & 1];
        int valid = n - c * CHUNK_ROWS;
        if (valid > CHUNK_ROWS) valid = CHUNK_ROWS;
        for (int r = rg; r < valid; r += 4)
            m = fmaxf(m, b[r * KDIM + feat]);

        __syncthreads();            // safe to overwrite this buffer next iter
    }

    // reduce 4 row-groups per feature
    red[t] = m;
    __syncthreads();
    if (t < KDIM) {
        float v = fmaxf(fmaxf(red[t], red[t + 64]),
                        fmaxf(red[t + 128], red[t + 192]));
        out[bag * KDIM + t] = v;
    }
}

// ---------------------------------------------------------------------------
// Host-side launcher
// ---------------------------------------------------------------------------
extern "C" void kernel_launch(void* const* d_in, const int* in_sizes, int n_in,
                              void* d_out, int out_size, void* d_ws, size_t ws_size,
                              hipStream_t stream) {
    const float* inter_pre = (const float*)d_in[0];   // [TOTAL, 64] f32
    const int*   bags_size = (const int*)d_in[1];     // [N_BAGS] i32
    const int    n_bags    = in_sizes[1];

    int* starts = (int*)d_ws;                         // n_bags+1 ints scratch

    scan_kernel<<<1, 1024, 0, stream>>>(bags_size, starts, n_bags);
    pool_kernel<<<n_bags, 256, 0, stream>>>(inter_pre, starts, (float*)d_out);
}